// CPCLoss_54271206752618
// MI455X (gfx1250) — compile-verified
//
#include <hip/hip_runtime.h>
#include <math.h>

// Problem constants (from reference): B=16, T=512, D=256, N_NEG=128, TEMP=0.07
#define B_    16
#define T_    512
#define D_    256
#define NNEG  128
#define BT    (B_ * T_)            // 8192 rows
#define GRIDX 511                  // ceil(max(B*L)/16) = 8176/16
#define NH    3
#define TEMP_INV 14.2857142857142857f   // 1/0.07

typedef __attribute__((ext_vector_type(2))) float v2f;
typedef __attribute__((ext_vector_type(8))) float v8f;

__device__ __forceinline__ float wave_reduce_sum(float v) {
    // full xor butterfly -> every lane ends with the wave32 sum
    v += __shfl_xor(v, 16, 32);
    v += __shfl_xor(v, 8, 32);
    v += __shfl_xor(v, 4, 32);
    v += __shfl_xor(v, 2, 32);
    v += __shfl_xor(v, 1, 32);
    return v;
}

// ---------------------------------------------------------------------------
// Kernel 1: all_z = normalize(z_seq) rows. One wave32 per row, 8 rows/block.
// ---------------------------------------------------------------------------
__global__ __launch_bounds__(256) void cpc_normalize_rows(
    const float* __restrict__ z, float* __restrict__ out) {
    const int wave = threadIdx.x >> 5;
    const int wl   = threadIdx.x & 31;
    const int row  = blockIdx.x * 8 + wave;   // grid = BT/8 blocks, always in range
    const float* src = z + (size_t)row * D_ + wl * 8;
    float4 a = *(const float4*)(src);
    float4 b = *(const float4*)(src + 4);
    float ss = a.x*a.x + a.y*a.y + a.z*a.z + a.w*a.w
             + b.x*b.x + b.y*b.y + b.z*b.z + b.w*b.w;
    ss = wave_reduce_sum(ss);
    const float inv = 1.0f / fmaxf(sqrtf(ss), 1e-12f);
    float* dst = out + (size_t)row * D_ + wl * 8;
    float4 oa = make_float4(a.x*inv, a.y*inv, a.z*inv, a.w*inv);
    float4 ob = make_float4(b.x*inv, b.y*inv, b.z*inv, b.w*inv);
    *(float4*)(dst)     = oa;
    *(float4*)(dst + 4) = ob;
}

// ---------------------------------------------------------------------------
// Kernel 2: fused per-horizon tile: fp32 WMMA GEMM (16 rows x 256 cols of
// z_pred) -> LDS -> normalize -> pos/neg logits -> online logsumexp -> partial
// grid = (GRIDX, NH), block = 256 (8 waves)
// ---------------------------------------------------------------------------
__global__ __launch_bounds__(256) void cpc_horizon_tile(
    const float* __restrict__ z, const float* __restrict__ preds,
    const int* __restrict__ neg_idx, const float* __restrict__ allz,
    float* __restrict__ partial) {

    __shared__ float zp[16][D_ + 4];   // +4 pad: conflict-free across lane halves
    __shared__ float wsum[8];

    const int h  = blockIdx.y;
    const int kH = (h == 0) ? 1 : (h == 1 ? 5 : 21);
    const int L  = T_ - kH;
    const int BL = B_ * L;
    const int rowbase = blockIdx.x * 16;
    if (rowbase >= BL) {                       // block-uniform early out
        if (threadIdx.x == 0) partial[h * GRIDX + blockIdx.x] = 0.0f;
        return;
    }

    const int wave = threadIdx.x >> 5;
    const int wl   = threadIdx.x & 31;
    const int lm   = wl & 15;                  // M index (A) / N index (B)
    const int half = wl >> 4;                  // K sub-block: 0 -> K{0,1}, 1 -> K{2,3}

    // ---- GEMM: z_pred[m, n] = sum_d z_seq[row(m), d] * preds[h][n, d] ----
    // A operand (16x4 f32): lane lm holds row m=lm, float2 at K = k0 + half*2
    int nA = rowbase + lm;
    int arow = 0;
    if (nA < BL) { int b = nA / L; int l = nA - b * L; arow = b * T_ + l; }
    const float* aP  = z + (size_t)arow * D_ + half * 2;
    // B operand (4x16 f32): lane lm holds column n, float2 over same K pair;
    // B[k][n] = preds[n][k] -> contiguous in memory along k.
    const int col0   = wave * 32 + lm;
    const float* bP0 = preds + (size_t)h * D_ * D_ + (size_t)col0 * D_ + half * 2;
    const float* bP1 = bP0 + 16 * D_;

    v8f acc0 = {};
    v8f acc1 = {};
    #pragma unroll 8
    for (int k0 = 0; k0 < D_; k0 += 4) {
        v2f a  = *(const v2f*)(aP  + k0);
        v2f b0 = *(const v2f*)(bP0 + k0);
        v2f b1 = *(const v2f*)(bP1 + k0);
        acc0 = __builtin_amdgcn_wmma_f32_16x16x4_f32(
            false, a, false, b0, (short)0, acc0, false, false);
        acc1 = __builtin_amdgcn_wmma_f32_16x16x4_f32(
            false, a, false, b1, (short)0, acc1, false, false);
    }

    // C/D layout: VGPR r -> row r (lanes 0-15) / row r+8 (lanes 16-31), col = lm
    #pragma unroll
    for (int r = 0; r < 8; ++r) {
        const int row = r + half * 8;
        zp[row][wave * 32 + lm]      = acc0[r];
        zp[row][wave * 32 + 16 + lm] = acc1[r];
    }
    __syncthreads();

    // ---- per-row logits + online logsumexp; wave handles 2 rows ----
    float rowAccum = 0.0f;
    #pragma unroll
    for (int rr = 0; rr < 2; ++rr) {
        const int m = wave * 2 + rr;
        const int n = rowbase + m;
        if (n >= BL) continue;                 // wave-uniform
        const int b = n / L;
        const int l = n - b * L;
        const int zsrc = b * T_ + l;

        float zr[8];
        float ss = 0.0f;
        #pragma unroll
        for (int j = 0; j < 8; ++j) { zr[j] = zp[m][wl + 32 * j]; ss += zr[j] * zr[j]; }
        ss = wave_reduce_sum(ss);
        const float inv = 1.0f / fmaxf(sqrtf(ss), 1e-12f);

        // positive logit
        const float* zpos = allz + (size_t)(zsrc + kH) * D_;
        float pd = 0.0f;
        #pragma unroll
        for (int j = 0; j < 8; ++j) pd += zr[j] * zpos[wl + 32 * j];
        pd = wave_reduce_sum(pd);
        const float pos_logit = pd * inv * TEMP_INV;

        // online logsumexp over [pos, 128 negatives]
        float mr = pos_logit;
        float sr = 1.0f;
        const int* nix = neg_idx + ((size_t)h * BT + n) * NNEG;
        for (int j = 0; j < NNEG; ++j) {
            const int idx = nix[j];
            const float* zn = allz + (size_t)idx * D_;
            float d = 0.0f;
            #pragma unroll
            for (int c = 0; c < 8; ++c) d += zr[c] * zn[wl + 32 * c];
            d = wave_reduce_sum(d);
            const float v = d * inv * TEMP_INV;
            if (v > mr) { sr = sr * __expf(mr - v) + 1.0f; mr = v; }
            else        { sr += __expf(v - mr); }
        }
        rowAccum += (mr + __logf(sr)) - pos_logit;
    }

    if (wl == 0) wsum[wave] = rowAccum;
    __syncthreads();
    if (threadIdx.x == 0) {
        float s = 0.0f;
        #pragma unroll
        for (int w = 0; w < 8; ++w) s += wsum[w];
        const float raw = 1.0f / sqrtf((float)kH);
        const float tot = 1.0f + 0.44721359549995793f + 0.21821789023599239f;
        partial[h * GRIDX + blockIdx.x] = s * (raw / tot) / (float)BL;
    }
}

// ---------------------------------------------------------------------------
// Kernel 3: deterministic reduction of partials into the scalar output.
// ---------------------------------------------------------------------------
__global__ __launch_bounds__(256) void cpc_reduce_partials(
    const float* __restrict__ partial, float* __restrict__ out, int n) {
    __shared__ float sm[256];
    float s = 0.0f;
    for (int i = threadIdx.x; i < n; i += 256) s += partial[i];
    sm[threadIdx.x] = s;
    __syncthreads();
    for (int st = 128; st > 0; st >>= 1) {
        if ((int)threadIdx.x < st) sm[threadIdx.x] += sm[threadIdx.x + st];
        __syncthreads();
    }
    if (threadIdx.x == 0) out[0] = sm[0];
}

extern "C" void kernel_launch(void* const* d_in, const int* in_sizes, int n_in,
                              void* d_out, int out_size, void* d_ws, size_t ws_size,
                              hipStream_t stream) {
    const float* z_seq   = (const float*)d_in[0];   // (16,512,256) f32
    const float* preds   = (const float*)d_in[1];   // (3,256,256)  f32
    const int*   neg_idx = (const int*)d_in[2];     // (3,8192,128) int

    float* allz    = (float*)d_ws;                    // BT*D floats (8 MB)
    float* partial = allz + (size_t)BT * D_;          // GRIDX*NH floats

    cpc_normalize_rows<<<BT / 8, 256, 0, stream>>>(z_seq, allz);

    dim3 grid(GRIDX, NH);
    cpc_horizon_tile<<<grid, 256, 0, stream>>>(z_seq, preds, neg_idx, allz, partial);

    cpc_reduce_partials<<<1, 256, 0, stream>>>(partial, (float*)d_out, GRIDX * NH);
}